// NodeBlock_81655918232105
// MI455X (gfx1250) — compile-verified
//
#include <hip/hip_runtime.h>

typedef __attribute__((ext_vector_type(2))) float v2f;
typedef __attribute__((ext_vector_type(8))) float v8f;

#define D 128
#define LN_EPS 1e-5f
#define WAVES_PER_BLOCK 4
#define LDS_ROW 132  // 128 + 4 pad -> conflict-free LDS banks

// ---------------- scatter-add (segment_sum via f32 atomics) ----------------
__global__ void scatter_add_kernel(const float* __restrict__ edge,
                                   const long long* __restrict__ recv,
                                   float* __restrict__ agg, int nE) {
  int t = blockIdx.x * blockDim.x + threadIdx.x;  // one thread per 4 floats
  int total = nE * (D / 4);
  if (t >= total) return;
  int e = t >> 5;         // 32 float4-chunks per edge row
  int c = (t & 31) << 2;  // column base
  const float4 v = *(const float4*)(edge + (size_t)e * D + c);
  float* dst = agg + (size_t)recv[e] * D + c;
  atomicAdd(dst + 0, v.x);
  atomicAdd(dst + 1, v.y);
  atomicAdd(dst + 2, v.z);
  atomicAdd(dst + 3, v.w);
}

// ------------- pack W1/W2 into WMMA B-fragment layout -----------------
// B fragment (16x16x4 f32): lane l, vgpr v holds B[K = (l>>4)*2 + v][N = l&15].
// Packed as float2 per lane: index ((tile*KSTEPS + kstep)*32 + lane).
__global__ void pack_w_kernel(const float* __restrict__ W1,
                              const float* __restrict__ W2,
                              v2f* __restrict__ W1p, v2f* __restrict__ W2p) {
  int t = blockIdx.x * blockDim.x + threadIdx.x;
  if (t < 8 * 96 * 32) {
    int lane = t & 31;
    int kstep = (t >> 5) % 96;
    int tile = t / (96 * 32);
    int k = kstep * 4 + ((lane >> 4) << 1);
    int n = tile * 16 + (lane & 15);
    v2f w;
    w.x = W1[(size_t)k * D + n];
    w.y = W1[(size_t)(k + 1) * D + n];
    W1p[t] = w;
  } else {
    int u = t - 8 * 96 * 32;
    if (u >= 8 * 32 * 32) return;
    int lane = u & 31;
    int kstep = (u >> 5) % 32;
    int tile = u / (32 * 32);
    int k = kstep * 4 + ((lane >> 4) << 1);
    int n = tile * 16 + (lane & 15);
    v2f w;
    w.x = W2[(size_t)k * D + n];
    w.y = W2[(size_t)(k + 1) * D + n];
    W2p[u] = w;
  }
}

// ------------- fused MLP + LayerNorm, one 16-row tile per wave -------------
__global__ __launch_bounds__(WAVES_PER_BLOCK * 32) void node_mlp_kernel(
    const float* __restrict__ x,
    const float* aggm,  // may alias `out` (per-tile rows only) -> no restrict
    const float* __restrict__ aggw, const v2f* __restrict__ W1p,
    const v2f* __restrict__ W2p, const float* __restrict__ b1,
    const float* __restrict__ b2, const float* __restrict__ gamma,
    const float* __restrict__ beta, float* out, int N) {
  __shared__ float h1s[WAVES_PER_BLOCK * 16 * LDS_ROW];

  const int lane = threadIdx.x & 31;
  const int wave = threadIdx.x >> 5;
  const int tile = blockIdx.x * WAVES_PER_BLOCK + wave;
  const int rbase = tile * 16;
  if (rbase >= N) return;  // whole-wave uniform exit

  const int lmod = lane & 15;
  const int lhalf = lane >> 4;
  int arow = rbase + lmod;
  if (arow >= N) arow = N - 1;  // clamp loads; stores are guarded below

  v8f acc[8];
#pragma unroll
  for (int t = 0; t < 8; ++t)
    acc[t] = (v8f){0.f, 0.f, 0.f, 0.f, 0.f, 0.f, 0.f, 0.f};

  // ----- Layer 1: [x | agg_mesh | agg_world] (16x384) @ W1 (384x128) -----
  const float* srcs[3] = {x, aggm, aggw};
  for (int s = 0; s < 3; ++s) {
    const float* A = srcs[s] + (size_t)arow * D + lhalf * 2;
    const int kbase = s * 32;
    for (int j = 0; j < 32; ++j) {
      v2f a = *(const v2f*)(A + j * 4);  // A[M=lmod][K=j*4+lhalf*2 +{0,1}]
      const int kg = kbase + j;
#pragma unroll
      for (int t = 0; t < 8; ++t) {
        v2f b = W1p[(size_t)(t * 96 + kg) * 32 + lane];
        acc[t] = __builtin_amdgcn_wmma_f32_16x16x4_f32(
            false, a, false, b, (short)0, acc[t], false, false);
      }
    }
  }

  // ----- bias + ReLU, stage h1 tile (16x128, padded) in LDS -----
  float* myh = h1s + wave * 16 * LDS_ROW;
#pragma unroll
  for (int t = 0; t < 8; ++t) {
    float bb = b1[t * 16 + lmod];
#pragma unroll
    for (int r = 0; r < 8; ++r) {
      float v = acc[t][r] + bb;  // element (M = r + lhalf*8, N = t*16 + lmod)
      v = v > 0.f ? v : 0.f;
      myh[(r + lhalf * 8) * LDS_ROW + t * 16 + lmod] = v;
    }
  }

  // ----- Layer 2: h1 (16x128) @ W2 (128x128) -----
#pragma unroll
  for (int t = 0; t < 8; ++t)
    acc[t] = (v8f){0.f, 0.f, 0.f, 0.f, 0.f, 0.f, 0.f, 0.f};
  const float* A2 = myh + lmod * LDS_ROW + lhalf * 2;
  for (int j = 0; j < 32; ++j) {
    v2f a = *(const v2f*)(A2 + j * 4);
#pragma unroll
    for (int t = 0; t < 8; ++t) {
      v2f b = W2p[(size_t)(t * 32 + j) * 32 + lane];
      acc[t] = __builtin_amdgcn_wmma_f32_16x16x4_f32(
          false, a, false, b, (short)0, acc[t], false, false);
    }
  }

  // ----- bias2 + LayerNorm (row = r + lhalf*8 lives in one 16-lane half) ---
  float gv[8], bv[8];
#pragma unroll
  for (int t = 0; t < 8; ++t) {
    float bb = b2[t * 16 + lmod];
#pragma unroll
    for (int r = 0; r < 8; ++r) acc[t][r] += bb;
    gv[t] = gamma[t * 16 + lmod];
    bv[t] = beta[t * 16 + lmod];
  }

#pragma unroll
  for (int r = 0; r < 8; ++r) {
    float s1 = 0.f, s2 = 0.f;
#pragma unroll
    for (int t = 0; t < 8; ++t) {
      float v = acc[t][r];
      s1 += v;
      s2 += v * v;
    }
    // reduce across the 16 lanes holding this row (xor masks stay in-half)
#pragma unroll
    for (int m = 1; m < 16; m <<= 1) {
      s1 += __shfl_xor(s1, m, 32);
      s2 += __shfl_xor(s2, m, 32);
    }
    const float mu = s1 * (1.f / (float)D);
    float var = s2 * (1.f / (float)D) - mu * mu;
    var = var > 0.f ? var : 0.f;
    const float rs = rsqrtf(var + LN_EPS);
    const int grow = rbase + r + lhalf * 8;
    if (grow < N) {
#pragma unroll
      for (int t = 0; t < 8; ++t) {
        const int n = t * 16 + lmod;
        out[(size_t)grow * D + n] = (acc[t][r] - mu) * rs * gv[t] + bv[t];
      }
    }
  }
}

extern "C" void kernel_launch(void* const* d_in, const int* in_sizes, int n_in,
                              void* d_out, int out_size, void* d_ws,
                              size_t ws_size, hipStream_t stream) {
  const float* node = (const float*)d_in[0];
  const float* edge = (const float*)d_in[1];
  const float* edgew = (const float*)d_in[2];
  const long long* recv = (const long long*)d_in[3];   // int64 in reference
  const long long* recvw = (const long long*)d_in[4];  // int64 in reference
  const float* W1 = (const float*)d_in[5];
  const float* b1 = (const float*)d_in[6];
  const float* W2 = (const float*)d_in[7];
  const float* b2 = (const float*)d_in[8];
  const float* gamma = (const float*)d_in[9];
  const float* beta = (const float*)d_in[10];

  const int N = in_sizes[0] / D;
  const int E = in_sizes[3];
  const int Ew = in_sizes[4];

  // agg_mesh accumulates directly in d_out (fully overwritten by final MLP;
  // each wave reads only its own 16 rows before writing them).
  float* aggm = (float*)d_out;
  float* aggw = (float*)d_ws;
  v2f* W1p = (v2f*)(aggw + (size_t)N * D);
  v2f* W2p = W1p + 8 * 96 * 32;

  hipMemsetAsync(d_out, 0, (size_t)N * D * sizeof(float), stream);
  hipMemsetAsync(d_ws, 0, (size_t)N * D * sizeof(float), stream);

  {
    int total = E * (D / 4);
    scatter_add_kernel<<<(total + 255) / 256, 256, 0, stream>>>(edge, recv,
                                                                aggm, E);
  }
  {
    int total = Ew * (D / 4);
    scatter_add_kernel<<<(total + 255) / 256, 256, 0, stream>>>(edgew, recvw,
                                                                aggw, Ew);
  }
  {
    int total = 8 * 96 * 32 + 8 * 32 * 32;
    pack_w_kernel<<<(total + 255) / 256, 256, 0, stream>>>(W1, W2, W1p, W2p);
  }
  {
    int tiles = (N + 15) / 16;
    int blocks = (tiles + WAVES_PER_BLOCK - 1) / WAVES_PER_BLOCK;
    node_mlp_kernel<<<blocks, WAVES_PER_BLOCK * 32, 0, stream>>>(
        node, aggm, aggw, W1p, W2p, b1, b2, gamma, beta, (float*)d_out, N);
  }
}